// SimpleCarDynamics_55576876810955
// MI455X (gfx1250) — compile-verified
//
#include <hip/hip_runtime.h>
#include <cstdint>
#include <cstddef>

// ---------------------------------------------------------------------------
// SimpleCarDynamics rollout for MI455X (gfx1250, wave32).
// One wave32 per trajectory k (16384 waves). T=128 handled as 4 chunks of 32
// (one timestep per lane). Prefix sums = ds_bpermute wave scans + sgpr carry.
// State row (8704B) staged global->LDS with global_load_async_to_lds_b128
// (th:TH_LOAD_NT: 142.6MB stream must not evict the ~1MB BEV tables from L2);
// output row assembled in LDS and written with global_store_async_from_lds_b128
// (th:TH_STORE_NT: written once, never re-read on device).
// ---------------------------------------------------------------------------

constexpr int T_STEPS    = 128;
constexpr int CH         = 17;
constexpr int ROW_FLOATS = T_STEPS * CH;   // 2176 floats = 8704 B = 17 * 512 B
constexpr float DT_F     = 0.02f;
constexpr float INV_DT   = 50.0f;          // 1/DT
constexpr float BWDT     = 0.08f;          // BW*DT
constexpr float INV_BWDT = 12.5f;          // 1/(BW*DT)

typedef float v2f __attribute__((ext_vector_type(2)));

__device__ __forceinline__ float bperm_f(int src_lane, float v) {
  return __int_as_float(__builtin_amdgcn_ds_bpermute(src_lane << 2, __float_as_int(v)));
}
__device__ __forceinline__ float lane31_f(float v) {
  return __int_as_float(__builtin_amdgcn_readlane(__float_as_int(v), 31));
}
// wave32 inclusive scan (Hillis-Steele, 5 ds_bpermute steps)
__device__ __forceinline__ float incl_scan32(float v, int lane) {
#pragma unroll
  for (int off = 1; off < 32; off <<= 1) {
    float up = bperm_f((lane - off) & 31, v);
    v += (lane >= off) ? up : 0.0f;
  }
  return v;
}

__global__ __launch_bounds__(64) void car_rollout_kernel(
    const float* __restrict__ state,   // (1,K,128,17)
    const float* __restrict__ pact,    // (K,128,2)
    const float* __restrict__ bevh,    // (256,256)
    const float* __restrict__ bevn,    // (256,256,3)
    float* __restrict__ ostate,        // (1,K,128,17)
    float* __restrict__ opa,           // (K,128,2)
    int K)
{
  __shared__ __align__(16) float smem[4 * ROW_FLOATS];  // 2 waves x (in + out)
  const int lane = threadIdx.x & 31;
  const int wv   = threadIdx.x >> 5;
  const int k    = blockIdx.x * 2 + wv;
  if (k >= K) return;  // K is a multiple of 2; guard is wave-uniform anyway

  float* lin  = smem + wv * ROW_FLOATS;
  float* lout = smem + (2 + wv) * ROW_FLOATS;
  // Low 32 bits of the generic (flat) address of a __shared__ object are the
  // wave-relative LDS byte offset.
  const unsigned lin_b  = (unsigned)(uintptr_t)(void*)lin;
  const unsigned lout_b = (unsigned)(uintptr_t)(void*)lout;

  // ---- Stage state row k -> LDS: 17 async b128 bursts (32 lanes x 16B each),
  //      non-temporal: read-once stream, keep L2 for the BEV gather tables.
  {
    unsigned long long g =
        (unsigned long long)(uintptr_t)(state + (size_t)k * ROW_FLOATS);
#pragma unroll
    for (int i = 0; i < 17; ++i) {
      unsigned la = lin_b + (unsigned)(i * 512 + lane * 16);
      unsigned long long ga = g + (unsigned long long)(i * 512 + lane * 16);
      asm volatile("global_load_async_to_lds_b128 %0, %1, off th:TH_LOAD_NT"
                   :: "v"(la), "v"(ga) : "memory");
    }
    asm volatile("s_wait_asynccnt 0x0" ::: "memory");
  }

  const float st12_0 = lin[12];  // state[k,0,12] (broadcast read, only t==0 uses)
  const float st13_0 = lin[13];  // state[k,0,13]

  float carry_s0 = 0.f, carry_s1 = 0.f;             // cumsum(pa) carries
  float carry_wz = 0.f, carry_x = 0.f, carry_y = 0.f;
  float prev_cd0 = 0.f, prev_cd1 = 0.f;             // last-lane values of prev chunk
  float prev_roll = 0.f, prev_pitch = 0.f;

#pragma unroll 1
  for (int cidx = 0; cidx < 4; ++cidx) {
    const int t = cidx * 32 + lane;

    // ---- controls = clip(ctrl0 + BW*DT*cumsum(pa))
    const v2f pav =
        __builtin_nontemporal_load((const v2f*)(pact + ((size_t)k * T_STEPS + t) * 2));
    float s0 = carry_s0 + incl_scan32(pav.x, lane);
    float s1 = carry_s1 + incl_scan32(pav.y, lane);
    carry_s0 = lane31_f(s0);
    carry_s1 = lane31_f(s1);

    const float st0  = lin[t * CH + 0];
    const float st1  = lin[t * CH + 1];
    const float st5  = lin[t * CH + 5];
    const float st9  = lin[t * CH + 9];
    const float st15 = lin[t * CH + 15];
    const float st16 = lin[t * CH + 16];

    const float c0 = fminf(fmaxf(st15 + BWDT * s0, -1.f), 1.f);
    const float c1 = fminf(fmaxf(st16 + BWDT * s1, 0.f), 0.5f);  // [0,0.5] within [-1,1]
    const float cd0 = c0 - st15;
    const float cd1 = c1 - st16;

    // ---- new_pa: t==0 -> pa[k,0]; else diff(controls-ctrl0)/(DT*BW)
    const float u0 = bperm_f((lane - 1) & 31, cd0);
    const float u1 = bperm_f((lane - 1) & 31, cd1);
    const float p0 = (lane == 0) ? prev_cd0 : u0;
    const float p1 = (lane == 0) ? prev_cd1 : u1;
    const float np0 = (t == 0) ? pav.x : (cd0 - p0) * INV_BWDT;
    const float np1 = (t == 0) ? pav.y : (cd1 - p1) * INV_BWDT;
    prev_cd0 = lane31_f(cd0);
    prev_cd1 = lane31_f(cd1);
    v2f npv; npv.x = np0; npv.y = np1;
    __builtin_nontemporal_store(npv, (v2f*)(opa + ((size_t)k * T_STEPS + t) * 2));

    // ---- kinematics
    const float steer    = c0;
    const float throttle = c1;
    const float vx = 17.0f * throttle;                 // SPEED_MAX * throttle
    const float dS = vx * DT_F;
    const float Kc = 2.0f * tanf(steer);               // tan(steer)/WHEELBASE(0.5)
    const float ayc = fminf(fmaxf(vx * (vx * Kc), -14.f), 14.f);
    const float wz  = ayc / fmaxf(vx, 1.0f);           // vx<=8.5 < 25

    float swz = carry_wz + incl_scan32(wz, lane);
    carry_wz = lane31_f(swz);
    const float yaw = st5 + DT_F * swz;
    float sy, cy;
    sincosf(yaw, &sy, &cy);

    float sxs = carry_x + incl_scan32(dS * cy, lane);
    carry_x = lane31_f(sxs);
    float sys = carry_y + incl_scan32(dS * sy, lane);
    carry_y = lane31_f(sys);
    const float x = st0 + sxs;
    const float y = st1 + sys;

    // ---- BEV lookups (RT policy -> stay L2/WGP$-resident);
    //      int cast truncates like astype(int32)
    int ix = (int)((x + 32.f) * 4.f); ix = ix < 0 ? 0 : (ix > 255 ? 255 : ix);
    int iy = (int)((y + 32.f) * 4.f); iy = iy < 0 ? 0 : (iy > 255 ? 255 : iy);
    const int gidx = iy * 256 + ix;
    const float z  = bevh[gidx];
    const float nx = bevn[gidx * 3 + 0];
    const float ny = bevn[gidx * 3 + 1];
    const float nz = bevn[gidx * 3 + 2];

    // left=cross(n,h), fwd=cross(left,n); only z-components needed
    const float roll  = asinf(nx * sy - ny * cy);
    const float pitch = asinf(nz * (nx * cy + ny * sy));

    const float ur  = bperm_f((lane - 1) & 31, roll);
    const float upt = bperm_f((lane - 1) & 31, pitch);
    const float pr = (lane == 0) ? prev_roll  : ur;
    const float pp = (lane == 0) ? prev_pitch : upt;
    const float wx = (t == 0) ? st12_0 : (roll  - pr) * INV_DT;
    const float wy = (t == 0) ? st13_0 : (pitch - pp) * INV_DT;
    prev_roll  = lane31_f(roll);
    prev_pitch = lane31_f(pitch);

    const float ax  = st9;
    const float ayo = vx * wz + 9.8f * sinf(roll);
    const float az  = -vx * wy + 9.8f * nz;

    // ---- assemble 17-channel output row in LDS (stride 17 dwords: bank-free)
    float* o = lout + t * CH;
    o[0] = x;    o[1] = y;    o[2] = z;    o[3] = roll;  o[4] = pitch;
    o[5] = yaw;  o[6] = vx;   o[7] = 0.f;  o[8] = 0.f;   o[9] = ax;
    o[10] = ayo; o[11] = az;  o[12] = wx;  o[13] = wy;   o[14] = wz;
    o[15] = steer; o[16] = throttle;
  }

  // ---- Coalesced async write-back of the output row (LDS -> global, NT)
  asm volatile("s_wait_dscnt 0x0" ::: "memory");  // DS stores visible before async read
  {
    unsigned long long g =
        (unsigned long long)(uintptr_t)(ostate + (size_t)k * ROW_FLOATS);
#pragma unroll
    for (int i = 0; i < 17; ++i) {
      unsigned la = lout_b + (unsigned)(i * 512 + lane * 16);
      unsigned long long ga = g + (unsigned long long)(i * 512 + lane * 16);
      asm volatile("global_store_async_from_lds_b128 %0, %1, off th:TH_STORE_NT"
                   :: "v"(ga), "v"(la) : "memory");
    }
    asm volatile("s_wait_asynccnt 0x0" ::: "memory");
  }
}

extern "C" void kernel_launch(void* const* d_in, const int* in_sizes, int n_in,
                              void* d_out, int out_size, void* d_ws, size_t ws_size,
                              hipStream_t stream) {
  (void)n_in; (void)out_size; (void)d_ws; (void)ws_size;
  const float* state = (const float*)d_in[0];   // (1,K,128,17) f32
  const float* pact  = (const float*)d_in[1];   // (K,128,2)    f32
  const float* bevh  = (const float*)d_in[2];   // (256,256)    f32
  const float* bevn  = (const float*)d_in[3];   // (256,256,3)  f32

  const int K = in_sizes[1] / (T_STEPS * 2);    // 16384

  float* out_states = (float*)d_out;                        // (1,K,128,17)
  float* out_pa     = out_states + (size_t)K * ROW_FLOATS;  // (K,128,2)

  const int waves_per_block = 2;                // 64 threads, 34.8 KB LDS/WG
  dim3 grid((K + waves_per_block - 1) / waves_per_block);
  dim3 block(64);
  car_rollout_kernel<<<grid, block, 0, stream>>>(state, pact, bevh, bevn,
                                                 out_states, out_pa, K);
}